// backBone_2980707304113
// MI455X (gfx1250) — compile-verified
//
#include <hip/hip_runtime.h>
#include <math.h>
#include <stdint.h>

typedef __attribute__((ext_vector_type(2))) float v2f;
typedef __attribute__((ext_vector_type(8))) float v8f;

#define BATCH 8
#define NPTS  4096
#define EPSBN 1e-5f
#define NSLOPE 0.2f

// ---------------- workspace layout (floats) ----------------
#define OFF_MU    0         // mu[24]
#define OFF_MAX   24        // uint max bits
#define OFF_S2    25        // 1/max
#define OFF_SC1   32
#define OFF_SH1   48
#define OFF_SC2   64
#define OFF_SH2   128
#define OFF_SC3   192
#define OFF_SH3   448
#define OFF_SC4   704
#define OFF_SH4   832
#define OFF_XC    1024      // xc4 [B*4*N] = 131072 (channel 3 zero-padded)
#define OFF_XX0   132096    // xx0 [B*N]   = 32768
#define OFF_F     164864    // f   [B*3*N] = 98304
#define OFF_Y1    263168    // y1  [B*16*N]
#define OFF_Y2    787456    // y2  [B*64*N]
#define OFF_Y3    2884608   // y3  [B*256*N]  (ends at 11273216 floats ~ 45 MB)

// -------- per-(b,channel) mean over N points --------
__global__ void k_mean(const float* __restrict__ x, float* __restrict__ mu) {
    __shared__ float s[256];
    const int bc = blockIdx.x;                 // b*3+c
    const float* p = x + (size_t)bc * NPTS;
    float acc = 0.f;
    for (int n = threadIdx.x; n < NPTS; n += 256) acc += p[n];
    s[threadIdx.x] = acc;
    __syncthreads();
    for (int w = 128; w > 0; w >>= 1) {
        if (threadIdx.x < w) s[threadIdx.x] += s[threadIdx.x + w];
        __syncthreads();
    }
    if (threadIdx.x == 0) mu[bc] = s[0] * (1.0f / NPTS);
}

__global__ void k_init(float* ws) {
    ((unsigned int*)ws)[OFF_MAX] = 0u;
}

// -------- center into padded [B][4][N] layout, squared norms, global max --------
__global__ void k_center(const float* __restrict__ x, const float* __restrict__ mu,
                         float* __restrict__ xc4, float* __restrict__ xx0,
                         unsigned int* __restrict__ maxbits) {
    __shared__ float s[256];
    const int id = blockIdx.x * 256 + threadIdx.x;   // b*N + n
    const int b = id / NPTS, n = id % NPTS;
    const size_t src = (size_t)b * 3 * NPTS + n;
    const size_t dst = (size_t)b * 4 * NPTS + n;
    float c0 = x[src          ] - mu[b * 3 + 0];
    float c1 = x[src + NPTS   ] - mu[b * 3 + 1];
    float c2 = x[src + 2*NPTS ] - mu[b * 3 + 2];
    xc4[dst          ] = c0;
    xc4[dst + NPTS   ] = c1;
    xc4[dst + 2*NPTS ] = c2;
    xc4[dst + 3*NPTS ] = 0.0f;                       // K-pad channel
    float xx = c0*c0 + c1*c1 + c2*c2;
    xx0[id] = xx;
    s[threadIdx.x] = xx;
    __syncthreads();
    for (int w = 128; w > 0; w >>= 1) {
        if (threadIdx.x < w) s[threadIdx.x] = fmaxf(s[threadIdx.x], s[threadIdx.x + w]);
        __syncthreads();
    }
    if (threadIdx.x == 0) atomicMax(maxbits, __float_as_uint(s[0]));  // xx>=0
}

__global__ void k_scale(float* ws) {
    ws[OFF_S2] = 1.0f / __uint_as_float(((unsigned int*)ws)[OFF_MAX]);
}

// -------- pairwise rowwise max/argmax of (xx_j - 2 x_i.x_j) --------
// 8 waves per block; batch point set staged to LDS via async global->LDS copies;
// each wave owns one 16-row tile and streams all 256 column tiles from LDS
// through V_WMMA_F32_16X16X4_F32.
__global__ __launch_bounds__(256) void k_pairwise(const float* __restrict__ xc4,
                                                  const float* __restrict__ xx0,
                                                  const float* __restrict__ s2p,
                                                  float* __restrict__ f) {
    __shared__ float smem[5 * NPTS];                 // [0,4N): coords, [4N,5N): xx
    const int b   = blockIdx.y;
    const int tid = threadIdx.x;

    // ---- async stage: 4096 + 1024 x b128 transfers ----
    const unsigned sbase = (unsigned)(uintptr_t)smem;        // LDS byte offset
    const float* gxc = xc4 + (size_t)b * 4 * NPTS;
    for (int i = tid; i < NPTS; i += 256) {                  // 4*N floats = N float4
        unsigned laddr = sbase + (unsigned)i * 16u;
        const float* gp = gxc + (size_t)i * 4;
        asm volatile("global_load_async_to_lds_b128 %0, %1, off"
                     :: "v"(laddr), "v"(gp) : "memory");
    }
    const float* gxx = xx0 + (size_t)b * NPTS;
    for (int i = tid; i < NPTS / 4; i += 256) {
        unsigned laddr = sbase + 4u * NPTS * 4u + (unsigned)i * 16u;
        const float* gp = gxx + (size_t)i * 4;
        asm volatile("global_load_async_to_lds_b128 %0, %1, off"
                     :: "v"(laddr), "v"(gp) : "memory");
    }
    asm volatile("s_wait_asynccnt 0x0" ::: "memory");
    __syncthreads();

    const float* sxx = smem + 4 * NPTS;
    const int wave = tid >> 5, lane = tid & 31;
    const int l16 = lane & 15, hi = lane >> 4;
    const int rt  = blockIdx.x * 8 + wave;           // this wave's row tile
    const int pi  = rt * 16 + l16;

    v2f A;
    A[0] = smem[(2 * hi    ) * NPTS + pi];
    A[1] = smem[(2 * hi + 1) * NPTS + pi];           // channel 3 slab is zeros

    float best[8];
    int   bidx[8];
#pragma unroll
    for (int r = 0; r < 8; ++r) { best[r] = -3.4e38f; bidx[r] = 0; }

#pragma unroll 4
    for (int jt = 0; jt < NPTS / 16; ++jt) {
        const int q = jt * 16 + l16;
        v2f Bm;
        Bm[0] = smem[(2 * hi    ) * NPTS + q];
        Bm[1] = smem[(2 * hi + 1) * NPTS + q];
        v8f c = {};
        v8f g = __builtin_amdgcn_wmma_f32_16x16x4_f32(false, A, false, Bm,
                                                      (short)0, c, false, false);
        const float xxj = sxx[q];
#pragma unroll
        for (int r = 0; r < 8; ++r) {
            float v = xxj - 2.0f * g[r];
            if (v > best[r]) { best[r] = v; bidx[r] = q; }
        }
    }

    // max/argmax across the 16 lanes of each half (rows r / r+8)
#pragma unroll
    for (int r = 0; r < 8; ++r) {
        for (int off = 8; off >= 1; off >>= 1) {
            float ov = __shfl_xor(best[r], off, 32);
            int   oi = __shfl_xor(bidx[r], off, 32);
            if (ov > best[r] || (ov == best[r] && oi < bidx[r])) {
                best[r] = ov; bidx[r] = oi;
            }
        }
    }

    const float s2 = s2p[0];
    if (l16 == 0) {
        float* fb = f + (size_t)b * 3 * NPTS;
#pragma unroll
        for (int r = 0; r < 8; ++r) {
            const int row = rt * 16 + r + hi * 8;
            const float xxi = sxx[row];
            fb[row          ] = s2 * xxi;                 // xx (scaled)
            fb[row + NPTS   ] = s2 * sxx[bidx[r]];        // ||furthest||^2
            fb[row + 2*NPTS ] = s2 * (xxi + best[r]);     // furthest distance
        }
    }
}

// -------- channel GEMM via WMMA f32 16x16x4, fused input BN-norm + leaky relu --------
template <int C, bool NORM>
__global__ __launch_bounds__(32) void k_gemm(const float* __restrict__ W,
                                             const float* __restrict__ H,
                                             float* __restrict__ Y,
                                             const float* __restrict__ isc,
                                             const float* __restrict__ ish, int O) {
    const int mb   = blockIdx.x * 16;
    const int nb   = blockIdx.y * 16;
    const int b    = blockIdx.z;
    const int lane = threadIdx.x;
    const int l16  = lane & 15;
    const int hi   = lane >> 4;
    const float* Hb = H + (size_t)b * C * NPTS;
    constexpr int KT = (C + 3) / 4;

    v8f acc = {};
#pragma unroll 4
    for (int kt = 0; kt < KT; ++kt) {
        int k0 = kt * 4 + hi * 2;
        int k1 = k0 + 1;
        float m0 = 1.0f, m1 = 1.0f;
        if constexpr (C % 4 != 0) {                  // clamp+mask: no EXEC divergence
            if (k0 >= C) { k0 = 0; m0 = 0.0f; }
            if (k1 >= C) { k1 = 0; m1 = 0.0f; }
        }
        const int row = mb + l16;
        v2f A, Bm;
        A[0] = W[row * C + k0] * m0;
        A[1] = W[row * C + k1] * m1;
        float h0 = Hb[(size_t)k0 * NPTS + nb + l16];
        float h1 = Hb[(size_t)k1 * NPTS + nb + l16];
        if (NORM) {
            h0 = h0 * isc[k0] + ish[k0];
            h0 = h0 >= 0.0f ? h0 : NSLOPE * h0;
            h1 = h1 * isc[k1] + ish[k1];
            h1 = h1 >= 0.0f ? h1 : NSLOPE * h1;
        }
        Bm[0] = h0 * m0;
        Bm[1] = h1 * m1;
        acc = __builtin_amdgcn_wmma_f32_16x16x4_f32(false, A, false, Bm,
                                                    (short)0, acc, false, false);
        if (kt + 1 < KT)                              // gfx1250 global_prefetch_b8
            __builtin_prefetch(&Hb[(size_t)(kt * 4 + 4 + hi * 2) * NPTS + nb + l16], 0, 0);
    }
    float* Yb = Y + (size_t)b * O * NPTS;
#pragma unroll
    for (int r = 0; r < 8; ++r) {
        const int row = mb + r + hi * 8;
        Yb[(size_t)row * NPTS + nb + l16] = acc[r];
    }
}

// -------- per-channel BN stats over (B,N) -> scale/shift --------
__global__ void k_bnstats(const float* __restrict__ Y, const float* __restrict__ g,
                          const float* __restrict__ beta, float* __restrict__ sc,
                          float* __restrict__ sh, int O) {
    __shared__ float ssum[256], ssq[256];
    const int o = blockIdx.x;
    float s = 0.f, q = 0.f;
    for (int i = threadIdx.x; i < BATCH * NPTS; i += 256) {
        const int b = i / NPTS, n = i % NPTS;
        float v = Y[((size_t)b * O + o) * NPTS + n];
        s += v; q += v * v;
    }
    ssum[threadIdx.x] = s; ssq[threadIdx.x] = q;
    __syncthreads();
    for (int w = 128; w > 0; w >>= 1) {
        if (threadIdx.x < w) {
            ssum[threadIdx.x] += ssum[threadIdx.x + w];
            ssq[threadIdx.x]  += ssq[threadIdx.x + w];
        }
        __syncthreads();
    }
    if (threadIdx.x == 0) {
        const float inv = 1.0f / (BATCH * NPTS);
        const float mean = ssum[0] * inv;
        const float var  = ssq[0] * inv - mean * mean;
        const float scl  = g[o] * rsqrtf(var + EPSBN);
        sc[o] = scl;
        sh[o] = beta[o] - mean * scl;
    }
}

// -------- final in-place normalize + leaky relu on d_out --------
__global__ void k_norm_out(float* __restrict__ Y, const float* __restrict__ sc,
                           const float* __restrict__ sh) {
    const size_t e = (size_t)blockIdx.x * 256 + threadIdx.x;   // B*128*N elems
    const int o = (int)((e / NPTS) % 128);
    float v = Y[e] * sc[o] + sh[o];
    Y[e] = v >= 0.0f ? v : NSLOPE * v;
}

extern "C" void kernel_launch(void* const* d_in, const int* in_sizes, int n_in,
                              void* d_out, int out_size, void* d_ws, size_t ws_size,
                              hipStream_t stream) {
    const float* x  = (const float*)d_in[0];
    const float* W1 = (const float*)d_in[1];
    const float* g1 = (const float*)d_in[2];
    const float* b1 = (const float*)d_in[3];
    const float* W2 = (const float*)d_in[4];
    const float* g2 = (const float*)d_in[5];
    const float* b2 = (const float*)d_in[6];
    const float* W3 = (const float*)d_in[7];
    const float* g3 = (const float*)d_in[8];
    const float* b3 = (const float*)d_in[9];
    const float* W4 = (const float*)d_in[10];
    const float* g4 = (const float*)d_in[11];
    const float* b4 = (const float*)d_in[12];
    float* out = (float*)d_out;
    float* ws  = (float*)d_ws;

    float* mu  = ws + OFF_MU;
    float* xc4 = ws + OFF_XC;
    float* xx0 = ws + OFF_XX0;
    float* f   = ws + OFF_F;
    float* y1  = ws + OFF_Y1;
    float* y2  = ws + OFF_Y2;
    float* y3  = ws + OFF_Y3;

    k_mean<<<dim3(BATCH * 3), dim3(256), 0, stream>>>(x, mu);
    k_init<<<1, 1, 0, stream>>>(ws);
    k_center<<<dim3(BATCH * NPTS / 256), dim3(256), 0, stream>>>(
        x, mu, xc4, xx0, (unsigned int*)ws + OFF_MAX);
    k_scale<<<1, 1, 0, stream>>>(ws);
    k_pairwise<<<dim3(NPTS / 128, BATCH), dim3(256), 0, stream>>>(
        xc4, xx0, ws + OFF_S2, f);

    // layer 1: 3 -> 16 (raw features, no input norm)
    k_gemm<3, false><<<dim3(1, NPTS / 16, BATCH), dim3(32), 0, stream>>>(
        W1, f, y1, nullptr, nullptr, 16);
    k_bnstats<<<dim3(16), dim3(256), 0, stream>>>(y1, g1, b1, ws + OFF_SC1, ws + OFF_SH1, 16);
    // layer 2: 16 -> 64
    k_gemm<16, true><<<dim3(4, NPTS / 16, BATCH), dim3(32), 0, stream>>>(
        W2, y1, y2, ws + OFF_SC1, ws + OFF_SH1, 64);
    k_bnstats<<<dim3(64), dim3(256), 0, stream>>>(y2, g2, b2, ws + OFF_SC2, ws + OFF_SH2, 64);
    // layer 3: 64 -> 256
    k_gemm<64, true><<<dim3(16, NPTS / 16, BATCH), dim3(32), 0, stream>>>(
        W3, y2, y3, ws + OFF_SC2, ws + OFF_SH2, 256);
    k_bnstats<<<dim3(256), dim3(256), 0, stream>>>(y3, g3, b3, ws + OFF_SC3, ws + OFF_SH3, 256);
    // layer 4: 256 -> 128 (raw GEMM straight into d_out)
    k_gemm<256, true><<<dim3(8, NPTS / 16, BATCH), dim3(32), 0, stream>>>(
        W4, y3, out, ws + OFF_SC3, ws + OFF_SH3, 128);
    k_bnstats<<<dim3(128), dim3(256), 0, stream>>>(out, g4, b4, ws + OFF_SC4, ws + OFF_SH4, 128);
    k_norm_out<<<dim3(BATCH * 128 * NPTS / 256), dim3(256), 0, stream>>>(
        out, ws + OFF_SC4, ws + OFF_SH4);
}